// MyGCN_2860448219412
// MI455X (gfx1250) — compile-verified
//
#include <hip/hip_runtime.h>
#include <stdint.h>

typedef float v2f __attribute__((ext_vector_type(2)));
typedef float v8f __attribute__((ext_vector_type(8)));

#define N_NODES 100000
#define N_EDGES 3200000
#define DCH 32

// ---------------- degree / normalization ----------------

__global__ void gcn_init_deg(uint32_t* __restrict__ deg, int n) {
    int i = blockIdx.x * blockDim.x + threadIdx.x;
    if (i < n) deg[i] = 1u;  // self-loop
}

__global__ void gcn_count_deg(const long long* __restrict__ dst,
                              uint32_t* __restrict__ deg, int e) {
    int i = blockIdx.x * blockDim.x + threadIdx.x;
    if (i < e) atomicAdd(&deg[(int)dst[i]], 1u);
}

__global__ void gcn_make_dinv(const uint32_t* __restrict__ deg,
                              float* __restrict__ dinv, int n) {
    int i = blockIdx.x * blockDim.x + threadIdx.x;
    if (i < n) dinv[i] = rsqrtf((float)deg[i]);
}

// ---------------- fused dense transform via WMMA f32 16x16x4 ----------------
// H = act(X) @ W^T, where act(x) = RELU_IN ? relu(x + bin[k]) : x  (fuses the
// previous layer's bias+ReLU into the A-operand load; exact since every H
// element is consumed by exactly one tile's A operand).
// Epilogue additionally writes AGG = H * dinv[row]^2 (the self-loop term),
// removing the separate init pass over the aggregation buffer.
//
// f32 A 16x4 layout : lane&15 = M, lane>>4 selects K-half; VGPR0/1 = K pair.
// f32 B 4x16 layout : lane&15 = N, lane>>4 selects K-half; VGPR0/1 = K pair.
// f32 C/D 16x16     : VGPR r -> M = (lane>>4)*8 + r, N = lane&15.
template <bool RELU_IN>
__global__ __launch_bounds__(256)
void gcn_gemm_fused(const float* __restrict__ X, const float* __restrict__ W,
                    const float* __restrict__ bin, const float* __restrict__ dinv,
                    float* __restrict__ H, float* __restrict__ AGG, int n) {
    const int lane = threadIdx.x & 31;
    const int wave = blockIdx.x * (blockDim.x >> 5) + (threadIdx.x >> 5);
    const int row0 = wave * 16;
    if (row0 >= n) return;                 // whole-wave guard: EXEC all-ones for WMMA

    const int l16  = lane & 15;
    const int half = lane >> 4;            // 0 -> K={0,1}, 1 -> K={2,3} within a chunk
    const int arow = row0 + l16;           // A row for this lane
    const int koff = half * 2;

    v8f c0 = {};                           // output cols 0..15
    v8f c1 = {};                           // output cols 16..31

#pragma unroll
    for (int kc = 0; kc < 8; ++kc) {
        const int kbase = kc * 4 + koff;
        v2f a  = *reinterpret_cast<const v2f*>(X + (size_t)arow * DCH + kbase);
        if (RELU_IN) {
            a.x = fmaxf(a.x + bin[kbase],     0.0f);
            a.y = fmaxf(a.y + bin[kbase + 1], 0.0f);
        }
        v2f b0 = *reinterpret_cast<const v2f*>(W + (size_t)l16 * DCH + kbase);
        v2f b1 = *reinterpret_cast<const v2f*>(W + (size_t)(l16 + 16) * DCH + kbase);
        c0 = __builtin_amdgcn_wmma_f32_16x16x4_f32(false, a, false, b0,
                                                   (short)0, c0, false, false);
        c1 = __builtin_amdgcn_wmma_f32_16x16x4_f32(false, a, false, b1,
                                                   (short)0, c1, false, false);
    }

#pragma unroll
    for (int r = 0; r < 8; ++r) {
        const int m   = half * 8 + r;
        const int row = row0 + m;
        const float di = dinv[row];
        const float sc = di * di;
        float* rowp = H   + (size_t)row * DCH;
        float* aggp = AGG + (size_t)row * DCH;
        rowp[l16]      = c0[r];
        rowp[l16 + 16] = c1[r];
        aggp[l16]      = c0[r] * sc;       // self-loop contribution
        aggp[l16 + 16] = c1[r] * sc;
    }
}

// ---------------- edge scatter: AGG[dst] += H[src] * dinv[src]*dinv[dst] ----------------
// Thread = (edge, channel); the 32 lanes of a wave share one edge, so the
// index/dinv reads collapse to one request per wave and the H[src] row read is
// one coalesced 128B line. One global_atomic_add_f32 per lane (irreducible).
__global__ void gcn_edge_scatter(const long long* __restrict__ src,
                                 const long long* __restrict__ dst,
                                 const float* __restrict__ H,
                                 const float* __restrict__ dinv,
                                 float* __restrict__ AGG, int e) {
    long long gid = (long long)blockIdx.x * blockDim.x + threadIdx.x;
    if (gid < (long long)e * DCH) {
        int ei = (int)(gid >> 5);
        int ch = (int)(gid & 31);
        int s  = (int)src[ei];
        int t  = (int)dst[ei];
        float nm = dinv[s] * dinv[t];
        atomicAdd(&AGG[(size_t)t * DCH + ch], H[(size_t)s * DCH + ch] * nm);
    }
}

// ---------------- final epilogue: OUT = relu(AGG + b) ----------------
__global__ void gcn_bias_relu(const float* __restrict__ AGG,
                              const float* __restrict__ b,
                              float* __restrict__ OUT, int n) {
    int gid = blockIdx.x * blockDim.x + threadIdx.x;
    if (gid < n * DCH) {
        OUT[gid] = fmaxf(AGG[gid] + b[gid & 31], 0.0f);
    }
}

extern "C" void kernel_launch(void* const* d_in, const int* in_sizes, int n_in,
                              void* d_out, int out_size, void* d_ws, size_t ws_size,
                              hipStream_t stream) {
    const float*     x   = (const float*)d_in[0];
    const long long* ei  = (const long long*)d_in[1];   // int64 [2, E]
    const float*     W1  = (const float*)d_in[2];
    const float*     b1  = (const float*)d_in[3];
    const float*     W2  = (const float*)d_in[4];
    const float*     b2  = (const float*)d_in[5];
    float*           out = (float*)d_out;

    const long long* src = ei;
    const long long* dst = ei + N_EDGES;

    char* ws = (char*)d_ws;
    uint32_t* deg  = (uint32_t*)ws;                         // 400 KB
    float*    dinv = (float*)(ws + (512 << 10));            // 400 KB
    float*    bufA = (float*)(ws + (1 << 20));              // 12.8 MB (H)
    float*    bufB = (float*)(ws + (1 << 20) + (16 << 20)); // 12.8 MB (AGG layer 1)

    const int TPB      = 256;
    const int gN       = (N_NODES + TPB - 1) / TPB;
    const int gE       = (N_EDGES + TPB - 1) / TPB;
    const int gNC      = (N_NODES * DCH + TPB - 1) / TPB;
    const long long ec = (long long)N_EDGES * DCH;
    const int gEC      = (int)((ec + TPB - 1) / TPB);
    const int tiles    = (N_NODES + 15) / 16;
    const int gGemm    = (tiles + 7) / 8;                   // 8 waves / block

    // normalization
    gcn_init_deg  <<<gN, TPB, 0, stream>>>(deg, N_NODES);
    gcn_count_deg <<<gE, TPB, 0, stream>>>(dst, deg, N_EDGES);
    gcn_make_dinv <<<gN, TPB, 0, stream>>>(deg, dinv, N_NODES);

    // ---- layer 1: H1=x@W1^T, AGG1(self) fused; then edge scatter ----
    gcn_gemm_fused<false><<<gGemm, TPB, 0, stream>>>(x, W1, nullptr, dinv,
                                                     bufA, bufB, N_NODES);
    gcn_edge_scatter     <<<gEC,   TPB, 0, stream>>>(src, dst, bufA, dinv,
                                                     bufB, N_EDGES);
    // bufB now holds pre-activation agg1; bias1+ReLU is fused into layer-2 A-load.

    // ---- layer 2: H2=relu(agg1+b1)@W2^T, AGG2(self) written into d_out ----
    gcn_gemm_fused<true> <<<gGemm, TPB, 0, stream>>>(bufB, W2, b1, dinv,
                                                     bufA, out, N_NODES);
    gcn_edge_scatter     <<<gEC,   TPB, 0, stream>>>(src, dst, bufA, dinv,
                                                     out, N_EDGES);
    gcn_bias_relu        <<<gNC,   TPB, 0, stream>>>(out, b2, out, N_NODES);
}